// ContextGuidedTokenShift_54769422958605
// MI455X (gfx1250) — compile-verified
//
#include <hip/hip_runtime.h>

// ContextGuidedTokenShift for MI455X (gfx1250).
// B=8, H=W=128 (N=16384), C=576 fp32. out = w*shift(x) + (1-w)*x.
// Memory-bound: ~604 MB effective HBM traffic -> ~26 us at 23.3 TB/s.
// All addressing is 32-bit byte offsets off SGPR bases (tensor < 4 GB),
// 8 tokens/thread amortize index/slab math, gather rides the CDNA5 async
// global->LDS DMA path (ASYNCcnt), NT stores keep L2 for the 2x-read input.

typedef float v4f __attribute__((ext_vector_type(4)));

#define C4_TOT   144            // 576/4 float4s per token
#define TOK_B    2304           // bytes per token (144 * 16)
#define BLOCK    256
#define TPT      8              // tokens per thread (divides row length 128)

__global__ __launch_bounds__(BLOCK)
void cgts_kernel(const float* __restrict__ xin,
                 const float* __restrict__ wptr,
                 float* __restrict__ outp)
{
    __shared__ v4f smem[TPT][BLOCK];

    // idx enumerates (token_group, c4), c4 fastest -> coalesced lanes
    unsigned idx = blockIdx.x * BLOCK + threadIdx.x;
    unsigned tg  = idx / C4_TOT;
    unsigned c4  = idx - tg * C4_TOT;
    unsigned token0 = tg * TPT;          // 8 consecutive tokens, same image row

    // ---- per-thread (uniform over the 8 tokens) shift decode ----
    // slab widths over channel/16 units: 4x4, 8x2, 4x1  -> branchless decode
    unsigned unit = c4 >> 2;                       // channel/16, 0..35
    unsigned ge16 = (unit >= 16u) ? 1u : 0u;
    unsigned ge32 = (unit >= 32u) ? 1u : 0u;
    unsigned slab = (unit >> (2u - ge16 - ge32)) - (ge16 * 4u + ge32 * 16u);

    // (dy,dx) per slab as signed nibbles, slab0 in the low nibble.
    // OFFSETS: (0,1)(0,-1)(1,0)(-1,0)(0,2)(0,-2)(2,0)(-2,0)
    //          (1,1)(-1,-1)(1,-1)(-1,1)(2,2)(-2,-2)(2,-2)(-2,2)
    const unsigned long long DY_TAB = 0xE2E2F1F1E200F100ull;
    const unsigned long long DX_TAB = 0x2EE21FF100E200F1ull;
    unsigned sh = slab * 4u;
    int dy = ((int)(((unsigned)(DY_TAB >> sh) & 0xFu) << 28)) >> 28;
    int dx = ((int)(((unsigned)(DX_TAB >> sh) & 0xFu) << 28)) >> 28;

    int y  = (int)((token0 >> 7) & 127u);
    int x0 = (int)(token0 & 127u);
    bool inby = ((unsigned)(y - dy) < 128u);       // row bound, same for all 8
    unsigned doff = (unsigned)(-((dy << 7) + dx) * TOK_B); // res->src byte delta

    // 32-bit byte offset of (token0, c4) -- whole tensor is < 4 GB
    unsigned res_off = token0 * (unsigned)TOK_B + c4 * 16u;

    const char* xb = (const char*)xin;
    char*       ob = (char*)outp;
    unsigned long long xbase = (unsigned long long)(size_t)xb;

    unsigned lds0 = (unsigned)(size_t)&smem[0][threadIdx.x];

    // ---- issue 8 async DMA gathers (global -> this thread's LDS slots) ----
    float weff[TPT];
    float w = *wptr;                                // uniform scalar load
#pragma unroll
    for (int i = 0; i < TPT; ++i) {
        bool inb = inby && ((unsigned)(x0 + i - dx) < 128u);
        weff[i] = inb ? w : 0.0f;                   // zero-fill via weight
        // OOB lanes fetch their own residual (valid memory); value killed by weff=0
        unsigned voff = res_off + (unsigned)(i * TOK_B) + (inb ? doff : 0u);
        unsigned lo   = lds0 + (unsigned)(i * BLOCK * 16);
        asm volatile("global_load_async_to_lds_b128 %0, %1, %2"
                     :: "v"(lo), "v"(voff), "s"(xbase) : "memory");
    }

    // ---- residual stream: regular cached loads (stay hot in L2) ----
    v4f cur[TPT];
#pragma unroll
    for (int i = 0; i < TPT; ++i)
        cur[i] = *(const v4f*)(xb + (size_t)res_off + (size_t)(i * TOK_B));

    float omw = 1.0f - w;

    asm volatile("s_wait_asynccnt 0x0" ::: "memory");

#pragma unroll
    for (int i = 0; i < TPT; ++i) {
        v4f sv = smem[i][threadIdx.x];
        v4f r;
        r.x = fmaf(weff[i], sv.x, omw * cur[i].x);
        r.y = fmaf(weff[i], sv.y, omw * cur[i].y);
        r.z = fmaf(weff[i], sv.z, omw * cur[i].z);
        r.w = fmaf(weff[i], sv.w, omw * cur[i].w);
        __builtin_nontemporal_store(r, (v4f*)(ob + (size_t)res_off + (size_t)(i * TOK_B)));
    }
}

extern "C" void kernel_launch(void* const* d_in, const int* in_sizes, int n_in,
                              void* d_out, int out_size, void* d_ws, size_t ws_size,
                              hipStream_t stream)
{
    const float* x = (const float*)d_in[0];   // (8, 16384, 576) fp32
    const float* w = (const float*)d_in[1];   // scalar weight
    float* out = (float*)d_out;

    // out_size = 8*16384*576 = 75,497,472 floats
    unsigned total4  = (unsigned)(out_size / 4);        // 18,874,368 float4s
    unsigned threads = total4 / TPT;                    // 2,359,296
    unsigned grid    = threads / BLOCK;                 // 9,216 (exact)

    cgts_kernel<<<grid, BLOCK, 0, stream>>>(x, w, out);
}